// ProposalTargetLayer_6511170421105
// MI455X (gfx1250) — compile-verified
//
#include <hip/hip_runtime.h>
#include <hip/hip_bf16.h>
#include <stdint.h>

// ---------------------------------------------------------------------------
// ProposalTargetLayer for MI455X (gfx1250, wave32).
// K1: WMMA-tiled IoU (v_wmma_f32_16x16x4_f32 computes area_a[m]+area_b[n]
//     outer-sum tiles), GT boxes staged to LDS via the Tensor Data Mover.
//     Max/argmax is division-free (cross-multiplied compares); one true
//     division per ROI at the end.
// K2: single-block first-k selection + deltas/one-hot epilogue.
// ---------------------------------------------------------------------------

#define N_ROI        200000
#define N_GT         200
#define GT_PAD       208          // 13 tiles of 16
#define GT_TILES     13
#define ROI_TILES    (N_ROI / 16) // 12500
#define R_OUT        256
#define NUM_CLASSES  91
#define OUT_ELEMS    (R_OUT * 4 + R_OUT * NUM_CLASSES) // 24320

typedef float        v8f  __attribute__((ext_vector_type(8)));
typedef float        v2f  __attribute__((ext_vector_type(2)));
typedef unsigned int u32x4 __attribute__((ext_vector_type(4)));
typedef int          i32x8 __attribute__((ext_vector_type(8)));
typedef int          i32x4 __attribute__((ext_vector_type(4)));

// ---------------------------------------------------------------------------
// Kernel 1: per-ROI max-IoU + argmax over 200 GT boxes.
// Each wave owns a tile of 16 ROIs; 8 waves per block; GT boxes live in LDS.
// ---------------------------------------------------------------------------
__global__ __launch_bounds__(256) void iou_argmax_kernel(
    const float* __restrict__ rois,   // [N_ROI,4] y1,x1,y2,x2
    const float* __restrict__ gts,    // [N_GT,4]
    float* __restrict__ merged,       // [N_ROI] max IoU
    int*   __restrict__ best)         // [N_ROI] argmax gt
{
    __shared__ float gt_lds[GT_PAD * 4];
    __shared__ float garea[GT_PAD];

    // ---- Stage GT boxes (200*4 floats = 3200 B) into LDS via TDM ----------
    if (threadIdx.x == 0) {
        unsigned long long ga = (unsigned long long)(uintptr_t)gts;
        unsigned int lds_off = (unsigned int)(uintptr_t)(&gt_lds[0]); // low 32b of flat addr == LDS offset
        u32x4 g0;
        g0[0] = 1u;                                  // count=1, user descriptor
        g0[1] = lds_off;                             // lds_addr
        g0[2] = (unsigned int)(ga & 0xFFFFFFFFull);  // global_addr[95:64]
        g0[3] = (unsigned int)((ga >> 32) & 0x1FFFFFFull) | (2u << 30); // addr hi | type=2
        i32x8 g1;
        g1[0] = (int)(2u << 16);      // workgroup_mask=0, data_size=2 (4B)
        g1[1] = (int)(800u << 16);    // atomic_barrier_addr=0 | tensor_dim0[15:0]=800
        g1[2] = (int)(1u << 16);      // tensor_dim0[31:16]=0 | tensor_dim1[15:0]=1
        g1[3] = (int)(800u << 16);    // tensor_dim1[31:16]=0 | tile_dim0=800
        g1[4] = 1;                    // tile_dim1=1 | tile_dim2=0
        g1[5] = 800;                  // tensor_dim0_stride lo32
        g1[6] = 0;                    // stride hi | tensor_dim1_stride lo
        g1[7] = 0;
        i32x4 gz; gz[0] = 0; gz[1] = 0; gz[2] = 0; gz[3] = 0;
        i32x8 gz8; gz8[0] = 0; gz8[1] = 0; gz8[2] = 0; gz8[3] = 0;
        gz8[4] = 0; gz8[5] = 0; gz8[6] = 0; gz8[7] = 0;
        __builtin_amdgcn_tensor_load_to_lds(g0, g1, gz, gz, gz8, 0);
        __builtin_amdgcn_s_wait_tensorcnt(0);
    }
    __syncthreads();

    // Zero-pad GTs 200..207 and precompute areas.
    if (threadIdx.x < GT_PAD) {
        int j = threadIdx.x;
        if (j >= N_GT) {
            gt_lds[j * 4 + 0] = 0.f; gt_lds[j * 4 + 1] = 0.f;
            gt_lds[j * 4 + 2] = 0.f; gt_lds[j * 4 + 3] = 0.f;
        }
        garea[j] = (gt_lds[j * 4 + 2] - gt_lds[j * 4 + 0]) *
                   (gt_lds[j * 4 + 3] - gt_lds[j * 4 + 1]);
    }
    __syncthreads();

    const int lane   = threadIdx.x & 31;
    const int wid    = threadIdx.x >> 5;
    const bool hiH   = lane >= 16;
    const int col    = lane & 15;       // N index within 16-wide tile
    const int mofs   = hiH ? 8 : 0;     // C-layout: lanes 16-31 hold M=v+8
    const int gwave  = blockIdx.x * 8 + wid;
    const int nwave  = gridDim.x * 8;

    for (int tile = gwave; tile < ROI_TILES; tile += nwave) {
        const int rbase = tile * 16;
        // lanes 0-15 own ROI rows 0..15 (lanes 16-31 load duplicates for shuffles)
        const float4 myroi = ((const float4*)rois)[rbase + col];
        const float  myarea = (myroi.z - myroi.x) * (myroi.w - myroi.y);

        // A matrix 16x4 f32: lanes 0-15 hold {K0=area_m, K1=1}; lanes 16-31 hold {K2=0, K3=0}
        v2f A; A[0] = hiH ? 0.f : myarea; A[1] = hiH ? 0.f : 1.0f;

        // Gather the 8 ROI boxes this lane needs (rows M = v + mofs).
        float rb0[8], rb1[8], rb2[8], rb3[8];
        #pragma unroll
        for (int v = 0; v < 8; ++v) {
            int src = v + mofs;
            rb0[v] = __shfl(myroi.x, src, 32);
            rb1[v] = __shfl(myroi.y, src, 32);
            rb2[v] = __shfl(myroi.z, src, 32);
            rb3[v] = __shfl(myroi.w, src, 32);
        }

        // Division-free running max: iou = inter/denom with denom > 0, so
        // iou_new > iou_best  <=>  inter_new*denom_best > inter_best*denom_new.
        float bI[8], bD[8]; int bidx[8];
        #pragma unroll
        for (int v = 0; v < 8; ++v) { bI[v] = -1.f; bD[v] = 1.f; bidx[v] = 0; }

        for (int gt = 0; gt < GT_TILES; ++gt) {
            const int j = gt * 16 + col;
            const float gy1 = gt_lds[j * 4 + 0], gx1 = gt_lds[j * 4 + 1];
            const float gy2 = gt_lds[j * 4 + 2], gx2 = gt_lds[j * 4 + 3];
            const float ga  = garea[j];

            // B matrix 4x16 f32: K0 row = ones, K1 row = gt areas, K2/K3 = 0.
            v2f B; B[0] = hiH ? 0.f : 1.0f; B[1] = hiH ? 0.f : ga;
            v8f C = {};
            // D[v] = area_roi[M] + area_gt[N]  (the union's additive part)
            v8f D = __builtin_amdgcn_wmma_f32_16x16x4_f32(
                false, A, false, B, (short)0, C, false, false);

            #pragma unroll
            for (int v = 0; v < 8; ++v) {
                float iy1 = fmaxf(rb0[v], gy1);
                float ix1 = fmaxf(rb1[v], gx1);
                float iy2 = fminf(rb2[v], gy2);
                float ix2 = fminf(rb3[v], gx2);
                float inter = fmaxf(iy2 - iy1, 0.f) * fmaxf(ix2 - ix1, 0.f);
                float denom = D[v] - inter + 1e-7f;     // union + eps, > 0
                if (inter * bD[v] > bI[v] * denom) {    // strict >: earliest j wins
                    bI[v] = inter; bD[v] = denom; bidx[v] = j;
                }
            }
        }

        // Reduce over the 16 columns within each 16-lane half (tie -> lowest gt idx,
        // matching jnp.argmax first-occurrence semantics).
        #pragma unroll
        for (int v = 0; v < 8; ++v) {
            #pragma unroll
            for (int s = 1; s < 16; s <<= 1) {
                float oI = __shfl_xor(bI[v], s, 32);
                float oD = __shfl_xor(bD[v], s, 32);
                int   oi = __shfl_xor(bidx[v], s, 32);
                float lhs = oI * bD[v];
                float rhs = bI[v] * oD;
                if (lhs > rhs || (lhs == rhs && oi < bidx[v])) {
                    bI[v] = oI; bD[v] = oD; bidx[v] = oi;
                }
            }
            if (col == v) {                    // lane v -> row v, lane 16+v -> row v+8
                int row = rbase + v + mofs;
                merged[row] = bI[v] / bD[v];   // single true division per ROI
                best[row]   = bidx[v];
            }
        }
    }
}

// ---------------------------------------------------------------------------
// Kernel 2: deterministic first-k selection + deltas/one-hot epilogue.
// Single block, 1024 threads (32 waves).
// ---------------------------------------------------------------------------
__global__ __launch_bounds__(1024) void select_emit_kernel(
    const float* __restrict__ rois,
    const float* __restrict__ gts,
    const int*   __restrict__ labels,
    const float* __restrict__ merged,
    const int*   __restrict__ best,
    float*       __restrict__ out)     // [256*4 deltas][256*91 onehot]
{
    __shared__ int wsP[32], wsN[32];
    __shared__ int sharedCnt, runPos, runNeg, chunkP, chunkN;
    __shared__ int sel[R_OUT];

    const int tid  = threadIdx.x;
    const int lane = tid & 31;
    const int wid  = tid >> 5;
    const unsigned int lmask = (1u << lane) - 1u;

    // ---- phase 1: count positives ----
    int local = 0;
    for (int i = tid; i < N_ROI; i += 1024)
        local += (merged[i] > 0.5f) ? 1 : 0;
    if (tid == 0) { sharedCnt = 0; runPos = 0; runNeg = 0; }
    __syncthreads();
    atomicAdd(&sharedCnt, local);
    __syncthreads();
    const int total_pos = min(64, sharedCnt);
    const int negCap    = R_OUT - total_pos;

    // ---- phase 2: take-first-k selection in index order ----
    for (int base = 0; base < N_ROI; base += 1024) {
        const int i = base + tid;
        float m = (i < N_ROI) ? merged[i] : 0.f;
        const bool pos = (i < N_ROI) && (m > 0.5f);
        const bool neg = (i < N_ROI) && (m < 0.5f) && (m > 0.1f);

        unsigned int bm = (unsigned int)__ballot(pos);
        unsigned int bn = (unsigned int)__ballot(neg);
        const int prW = __popc(bm & lmask);
        const int nrW = __popc(bn & lmask);
        if (lane == 0) { wsP[wid] = __popc(bm); wsN[wid] = __popc(bn); }
        __syncthreads();
        if (tid == 0) {
            int aP = 0, aN = 0;
            for (int w = 0; w < 32; ++w) {
                int tP = wsP[w]; wsP[w] = aP; aP += tP;
                int tN = wsN[w]; wsN[w] = aN; aN += tN;
            }
            chunkP = aP; chunkN = aN;
        }
        __syncthreads();
        if (pos) {
            int slot = runPos + wsP[wid] + prW;
            if (slot < total_pos) sel[slot] = i;
        }
        if (neg) {
            int slot = runNeg + wsN[wid] + nrW;
            if (slot < negCap) sel[total_pos + slot] = i;
        }
        __syncthreads();
        if (tid == 0) { runPos += chunkP; runNeg += chunkN; }
        __syncthreads();
        if (runPos >= total_pos && runNeg >= negCap) break;
    }
    const int total_neg = min(negCap, runNeg);
    const int total_sel = total_pos + total_neg;

    // ---- phase 3: emit outputs (zeros for padding rows) ----
    for (int k = tid; k < OUT_ELEMS; k += 1024) out[k] = 0.f;
    __syncthreads();

    if (tid < R_OUT) {
        const int r = tid;
        if (r < total_sel) {
            const int  i      = sel[r];
            const bool is_pos = r < total_pos;
            if (is_pos) {
                const int g = best[i];
                const float ry1 = rois[i * 4 + 0], rx1 = rois[i * 4 + 1];
                const float ry2 = rois[i * 4 + 2], rx2 = rois[i * 4 + 3];
                const float gy1 = gts[g * 4 + 0], gx1 = gts[g * 4 + 1];
                const float gy2 = gts[g * 4 + 2], gx2 = gts[g * 4 + 3];
                const float bh = ry2 - ry1, bw = rx2 - rx1;
                const float bcy = ry1 + 0.5f * bh, bcx = rx1 + 0.5f * bw;
                const float gh = gy2 - gy1, gw = gx2 - gx1;
                const float gcy = gy1 + 0.5f * gh, gcx = gx1 + 0.5f * gw;
                const float bh_s = (bh == 0.f) ? 1.f : bh;
                const float bw_s = (bw == 0.f) ? 1.f : bw;
                const float gh_s = (gh <= 0.f) ? 1.f : gh;
                const float gw_s = (gw <= 0.f) ? 1.f : gw;
                const float dy = (gh == 0.f) ? 0.f : (gcy - bcy) / bh_s;
                const float dx = (gw == 0.f) ? 0.f : (gcx - bcx) / bw_s;
                const float dh = (gh == 0.f) ? 0.f : logf(gh_s / bh_s);
                const float dw = (gw == 0.f) ? 0.f : logf(gw_s / bw_s);
                out[r * 4 + 0] = dy / 0.1f;
                out[r * 4 + 1] = dx / 0.1f;
                out[r * 4 + 2] = dh / 0.2f;
                out[r * 4 + 3] = dw / 0.2f;
                const int label = labels[g];
                out[R_OUT * 4 + r * NUM_CLASSES + label] = 1.0f;
            } else {
                // negatives: zero gt box -> zero deltas; label 0 -> one-hot class 0
                out[R_OUT * 4 + r * NUM_CLASSES + 0] = 1.0f;
            }
        }
    }
}

// ---------------------------------------------------------------------------
extern "C" void kernel_launch(void* const* d_in, const int* in_sizes, int n_in,
                              void* d_out, int out_size, void* d_ws, size_t ws_size,
                              hipStream_t stream) {
    (void)in_sizes; (void)n_in; (void)out_size; (void)ws_size;
    // input order: img_shape(i32), roi_bboxes(f32), gt_boxes(f32), gt_labels(i32)
    const float* rois   = (const float*)d_in[1];
    const float* gts    = (const float*)d_in[2];
    const int*   labels = (const int*)d_in[3];

    float* merged = (float*)d_ws;                                  // 800 KB
    int*   best   = (int*)((char*)d_ws + (size_t)N_ROI * sizeof(float)); // 800 KB

    const int blocks = (ROI_TILES + 7) / 8;  // one 16-ROI tile per wave, 8 waves/block
    iou_argmax_kernel<<<blocks, 256, 0, stream>>>(rois, gts, merged, best);
    select_emit_kernel<<<1, 1024, 0, stream>>>(rois, gts, labels, merged, best,
                                               (float*)d_out);
}